// EnhancedSpikingNeuron_90314572300886
// MI455X (gfx1250) — compile-verified
//
#include <hip/hip_runtime.h>
#include <hip/hip_bf16.h>

// LIF spiking-neuron scan: spikes[b,t,d], sequential over T with soft reset.
// Pure HBM streaming (268 MB @ 23.3 TB/s ~= 11.5 us floor). Strategy: CDNA5
// async global->LDS copies (ASYNCcnt) in a quad-buffered ring with PARTIAL
// waits (s_wait_asynccnt 48) so ~3 chunks/WG (~12 MB machine-wide) stay in
// flight continuously -- no drain-to-zero bubbles at chunk boundaries.

namespace {

constexpr int kB = 16;
constexpr int kT = 2048;
constexpr int kD = 1024;
constexpr float kBeta   = 0.9f;
constexpr float kThresh = 1.0f;

constexpr int kWG       = 64;               // threads per workgroup (2 waves)
constexpr int kDBlk     = 64;               // d-columns per workgroup
constexpr int kTChunk   = 64;               // timesteps staged per LDS buffer
constexpr int kNChunk   = kT / kTChunk;     // 32 chunks
constexpr int kNBuf     = 4;                // LDS ring depth
constexpr int kRowBytes = kDBlk * 4;        // 256 B per staged row
constexpr int kBufBytes = kTChunk * kRowBytes;   // 16 KB per buffer
constexpr int kIssues   = kTChunk / 4;      // 16 async issues per wave per chunk

// Per-lane async copy of 16B global -> LDS (tracked by ASYNCcnt). GVS mode:
// vdst = LDS byte addr VGPR, vaddr = 32-bit offset VGPR, saddr = SGPR base.
// NT hint: stream is consumed exactly once (from LDS), keep it out of L2.
__device__ __forceinline__ void async_b128(unsigned lds_off,
                                           const float* sbase,
                                           unsigned goff) {
  asm volatile("global_load_async_to_lds_b128 %0, %1, %2 th:TH_LOAD_NT"
               :
               : "v"(lds_off), "v"(goff), "s"(sbase)
               : "memory");
}

// Partial wait: allow N async ops to remain outstanding (in-order completion
// guarantees the oldest (issued - N) ops have landed).
template <int N>
__device__ __forceinline__ void wait_async_le() {
  asm volatile("s_wait_asynccnt %0" :: "i"(N) : "memory");
}

} // namespace

__global__ __launch_bounds__(kWG)
void lif_scan_async(const float* __restrict__ x,
                    const float* __restrict__ homeo_i,
                    float* __restrict__ out) {
  __shared__ float tile[kNBuf][kTChunk][kDBlk];   // 64 KB ring

  const int tid  = threadIdx.x;
  const int lane = tid & 31;
  const int wave = tid >> 5;              // 0..1

  const int blk = blockIdx.x;             // 0..255
  const int b   = blk >> 4;               // 16 d-blocks per batch entry
  const int d0  = (blk & 15) * kDBlk;

  const float* sbase = x + (size_t)b * kT * kD;          // per-batch base (SGPR)
  float* obase = out + (size_t)b * kT * kD + d0 + tid;   // this thread's column

  // Copy-lane mapping: 32 lanes cover 2 rows x 256 B per issue.
  const unsigned colb = (unsigned)(lane & 15) * 16u;     // byte col within row
  const unsigned rsel = (unsigned)(lane >> 4);           // 0/1: row within pair
  // Low 32 bits of the generic address of a __shared__ object == LDS byte offset.
  const unsigned lds_base0 = (unsigned)(size_t)(&tile[0][0][0]);

  auto issue_chunk = [&](int c, int buf) {
    const unsigned row0 = (unsigned)(wave * 2) + rsel;   // 0..3
    unsigned g0 = ((unsigned)(c * kTChunk) + row0) * (unsigned)(kD * 4)
                  + (unsigned)(d0 * 4) + colb;
    unsigned l0 = lds_base0 + (unsigned)(buf * kBufBytes)
                  + row0 * (unsigned)kRowBytes + colb;
#pragma unroll
    for (int i = 0; i < kIssues; ++i) {
      async_b128(l0 + (unsigned)(i * 4 * kRowBytes),
                 sbase,
                 g0 + (unsigned)(i * 4 * kD * 4));
    }
  };

  // Prologue: stage chunks 0..2 (48 outstanding ops/wave, < 63 cap).
  issue_chunk(0, 0);
  issue_chunk(1, 1);
  issue_chunk(2, 2);

  const float h = homeo_i[0];
  float mem = 0.0f;

  for (int c = 0; c < kNChunk; ++c) {
    const int buf = c & (kNBuf - 1);

    // Keep the ring full, then wait only for the OLDEST chunk to land:
    // threshold = 16 * (#chunks issued after chunk c).
    if (c + 3 < kNChunk) {
      issue_chunk(c + 3, (c + 3) & (kNBuf - 1));
      wait_async_le<3 * kIssues>();      // chunks c+1..c+3 may stay in flight
    } else if (c + 2 < kNChunk) {
      wait_async_le<2 * kIssues>();
    } else if (c + 1 < kNChunk) {
      wait_async_le<1 * kIssues>();
    } else {
      wait_async_le<0>();
    }
    __syncthreads();   // both waves' DMA for chunk c now visible in LDS

    float* op = obase + (size_t)(c * kTChunk) * kD;
#pragma unroll
    for (int r = 0; r < kTChunk; ++r) {
      const float inp = tile[buf][r][tid] + h;           // bank-conflict-free
      mem = __builtin_fmaf(kBeta, mem, inp);             // leaky integrate
      const float s = (mem > kThresh) ? 1.0f : 0.0f;     // threshold spike
      mem -= s;                                          // soft (subtract) reset
      __builtin_nontemporal_store(s, op);                // streaming NT store
      op += kD;
    }

    // Buffer-reuse guard: chunk c+4 (issued next iteration) overwrites
    // tile[buf]; every wave must be done reading chunk c first.
    __syncthreads();
  }
}

extern "C" void kernel_launch(void* const* d_in, const int* in_sizes, int n_in,
                              void* d_out, int out_size, void* d_ws, size_t ws_size,
                              hipStream_t stream) {
  (void)in_sizes; (void)n_in; (void)out_size; (void)d_ws; (void)ws_size;
  const float* x     = (const float*)d_in[0];
  const float* homeo = (const float*)d_in[1];
  float* out = (float*)d_out;

  dim3 grid(kB * (kD / kDBlk));   // 256 workgroups
  dim3 block(kWG);                // 64 threads (2 waves, wave32)
  lif_scan_async<<<grid, block, 0, stream>>>(x, homeo, out);
}